// BaseSAE_37374805410395
// MI455X (gfx1250) — compile-verified
//
#include <hip/hip_runtime.h>
#include <math.h>

// Problem constants (from reference config)
constexpr int D      = 4096;   // D_DATA
constexpr int NCAND  = 8192;   // candidate rows
constexpr int NRESET = 128;    // outer iterations / rows of V

struct Ctrl { int pivot; int active; int n_succ; };

typedef __attribute__((ext_vector_type(2))) float v2f;
typedef __attribute__((ext_vector_type(8))) float v8f;

// ---------------------------------------------------------------- init
__global__ void init_ctrl_kernel(Ctrl* c) {
    c->pivot  = 0;
    c->active = 1;
    c->n_succ = NRESET;
}

__global__ void zero_out_kernel(float* out, int n) {
    int i = blockIdx.x * blockDim.x + threadIdx.x;
    if (i < n) out[i] = 0.0f;
}

// mags2[row] = ||X[row]||^2   (one block per row)
__global__ __launch_bounds__(256)
void row_norms_kernel(const float* __restrict__ X, float* __restrict__ mags2) {
    const int row = blockIdx.x;
    const int t   = threadIdx.x;
    const float4* xr = reinterpret_cast<const float4*>(X + (size_t)row * D);
    float s = 0.0f;
#pragma unroll
    for (int q = 0; q < 4; ++q) {
        float4 a = xr[t * 4 + q];
        s += a.x * a.x + a.y * a.y + a.z * a.z + a.w * a.w;
    }
    __shared__ float red[256];
    red[t] = s;
    __syncthreads();
    for (int st = 128; st > 0; st >>= 1) {
        if (t < st) red[t] += red[t + st];
        __syncthreads();
    }
    if (t == 0) mags2[row] = red[0];
}

// ---------------------------------------------------------------- argmax
__global__ __launch_bounds__(1024)
void argmax_kernel(const float* __restrict__ mags2, Ctrl* ctrl) {
    __shared__ float sval[1024];
    __shared__ int   sidx[1024];
    const int t = threadIdx.x;
    float best = -INFINITY;
    int   bidx = 0;
    for (int r = t; r < NCAND; r += 1024) {
        float m = mags2[r];
        if (m > best) { best = m; bidx = r; }   // strictly > keeps first occurrence
    }
    sval[t] = best; sidx[t] = bidx;
    __syncthreads();
    for (int s = 512; s > 0; s >>= 1) {
        if (t < s) {
            float ov = sval[t + s]; int oi = sidx[t + s];
            if (ov > sval[t] || (ov == sval[t] && oi < sidx[t])) {
                sval[t] = ov; sidx[t] = oi;
            }
        }
        __syncthreads();
    }
    if (t == 0) ctrl->pivot = sidx[0];
}

// ---------------------------------------------------------------- pivot MGS
// Single workgroup: v = x[pivot]; two MGS passes over V[0..i-1]
// (pass 1 == accumulated deflations, pass 2 == reference's explicit MGS),
// then normalize and write V[i]. Handles the early-break bookkeeping.
__global__ __launch_bounds__(256)
void pivot_mgs_kernel(const float* __restrict__ X, float* __restrict__ V,
                      Ctrl* ctrl, int i) {
    if (!ctrl->active) return;                 // uniform
    __shared__ float red[256];
    __shared__ float red2[256];
    const int t = threadIdx.x;
    const int p = ctrl->pivot;

    // each thread owns 16 contiguous elements: [t*16, t*16+16)
    float v[16];
    {
        const float4* s = reinterpret_cast<const float4*>(X + (size_t)p * D) + t * 4;
#pragma unroll
        for (int q = 0; q < 4; ++q) {
            float4 a = s[q];
            v[4 * q + 0] = a.x; v[4 * q + 1] = a.y;
            v[4 * q + 2] = a.z; v[4 * q + 3] = a.w;
        }
    }

    for (int rep = 0; rep < 2; ++rep) {
        for (int j = 0; j < i; ++j) {
            const float4* uj4 = reinterpret_cast<const float4*>(V + (size_t)j * D) + t * 4;
            float u[16];
            float p1 = 0.0f, p2 = 0.0f;
#pragma unroll
            for (int q = 0; q < 4; ++q) {
                float4 a = uj4[q];
                u[4 * q + 0] = a.x; u[4 * q + 1] = a.y;
                u[4 * q + 2] = a.z; u[4 * q + 3] = a.w;
            }
#pragma unroll
            for (int e = 0; e < 16; ++e) {
                p1 += u[e] * v[e];
                p2 += u[e] * u[e];
            }
            red[t] = p1; red2[t] = p2;
            __syncthreads();
            for (int s = 128; s > 0; s >>= 1) {
                if (t < s) { red[t] += red[t + s]; red2[t] += red2[t + s]; }
                __syncthreads();
            }
            const float c   = red[0];
            const float den = red2[0];
            __syncthreads();                   // before red[] is reused
            const float sc = c / fmaxf(den, 1e-30f);
#pragma unroll
            for (int e = 0; e < 16; ++e) v[e] -= sc * u[e];
        }
    }

    // ||v||
    float p1 = 0.0f;
#pragma unroll
    for (int e = 0; e < 16; ++e) p1 += v[e] * v[e];
    red[t] = p1;
    __syncthreads();
    for (int s = 128; s > 0; s >>= 1) {
        if (t < s) red[t] += red[t + s];
        __syncthreads();
    }
    const float vn = sqrtf(red[0]);
    const bool failed = vn < 1e-6f;            // EPS_BREAK
    if (!failed) {
        const float inv = 1.0f / fmaxf(vn, 1e-12f);   // EPS_NORM
        float4* o = reinterpret_cast<float4*>(V + (size_t)i * D) + t * 4;
#pragma unroll
        for (int q = 0; q < 4; ++q) {
            float4 a;
            a.x = v[4 * q + 0] * inv; a.y = v[4 * q + 1] * inv;
            a.z = v[4 * q + 2] * inv; a.w = v[4 * q + 3] * inv;
            o[q] = a;
        }
    } else if (t == 0) {
        ctrl->n_succ = i;                      // record break index once
        ctrl->active = 0;                      // freeze everything after
    }
}

// ---------------------------------------------------------------- deflation
// d[r] = X[r] . u  via V_WMMA_F32_16X16X4_F32 (u broadcast into all 16 B
// columns), then mags2[r] -= d[r]^2.  One block per 16-row tile; 8 waves
// split K = 4096; two WMMA accumulator chains per wave for latency hiding.
__global__ __launch_bounds__(256)
void deflate_wmma_kernel(const float* __restrict__ X, const float* __restrict__ V,
                         float* __restrict__ mags2, const Ctrl* ctrl, int i) {
    if (!ctrl->active) return;                 // uniform
    const float* u = V + (size_t)i * D;

    const int tile  = blockIdx.x;              // rows [tile*16, tile*16+16)
    const int wave  = threadIdx.x >> 5;        // 0..7
    const int lane  = threadIdx.x & 31;
    const int rowIn = lane & 15;
    const int kHalf = (lane >> 4) ? 2 : 0;     // A layout: lanes 0-15 -> K=0,1 ; 16-31 -> K=2,3
    const float* rowPtr = X + (size_t)(tile * 16 + rowIn) * D;

    const int k0 = wave * (D / 8);             // 512 K per wave
    v8f c0 = {};
    v8f c1 = {};
    for (int k = k0; k < k0 + D / 8; k += 8) {
        v2f a0 = *reinterpret_cast<const v2f*>(rowPtr + k + kHalf);
        v2f b0 = *reinterpret_cast<const v2f*>(u      + k + kHalf);
        v2f a1 = *reinterpret_cast<const v2f*>(rowPtr + k + 4 + kHalf);
        v2f b1 = *reinterpret_cast<const v2f*>(u      + k + 4 + kHalf);
        c0 = __builtin_amdgcn_wmma_f32_16x16x4_f32(false, a0, false, b0,
                                                   (short)0, c0, false, false);
        c1 = __builtin_amdgcn_wmma_f32_16x16x4_f32(false, a1, false, b1,
                                                   (short)0, c1, false, false);
    }

    // D layout: VGPR g, lanes 0-15 hold M=g, lanes 16-31 hold M=g+8 (all N
    // columns identical since B columns are identical).  Lanes 0 and 16 carry
    // the 16 row sums; reduce the 8 K-waves through LDS.
    __shared__ float rowsum[16];
    if (threadIdx.x < 16) rowsum[threadIdx.x] = 0.0f;
    __syncthreads();
    if ((lane & 15) == 0) {
        const int base = (lane == 0) ? 0 : 8;
#pragma unroll
        for (int g = 0; g < 8; ++g)
            atomicAdd(&rowsum[base + g], c0[g] + c1[g]);
    }
    __syncthreads();
    if (threadIdx.x < 16) {
        const int r = tile * 16 + threadIdx.x;
        const float d = rowsum[threadIdx.x];
        mags2[r] -= d * d;                     // incremental norm update
    }
}

// ---------------------------------------------------------------- finalize
__global__ void write_nsucc_kernel(float* out, const Ctrl* c) {
    out[NRESET * D] = (float)c->n_succ;
}

// ---------------------------------------------------------------- launch
extern "C" void kernel_launch(void* const* d_in, const int* in_sizes, int n_in,
                              void* d_out, int out_size, void* d_ws, size_t ws_size,
                              hipStream_t stream) {
    const float* X = (const float*)d_in[0];          // [NCAND, D] fp32, read-only
    float* V       = (float*)d_out;                  // [NRESET, D] fp32 (+ n_succ slot)
    float* mags2   = (float*)d_ws;                   // [NCAND]
    Ctrl*  ctrl    = (Ctrl*)((char*)d_ws + NCAND * sizeof(float));

    init_ctrl_kernel<<<1, 1, 0, stream>>>(ctrl);
    zero_out_kernel<<<(out_size + 255) / 256, 256, 0, stream>>>(V, out_size);
    row_norms_kernel<<<NCAND, 256, 0, stream>>>(X, mags2);

    for (int i = 0; i < NRESET; ++i) {
        argmax_kernel<<<1, 1024, 0, stream>>>(mags2, ctrl);
        pivot_mgs_kernel<<<1, 256, 0, stream>>>(X, V, ctrl, i);
        deflate_wmma_kernel<<<NCAND / 16, 256, 0, stream>>>(X, V, mags2, ctrl, i);
    }

    if (out_size > NRESET * D)
        write_nsucc_kernel<<<1, 1, 0, stream>>>(V, ctrl);
}